// MagicGAT_19189913879032
// MI455X (gfx1250) — compile-verified
//
#include <hip/hip_runtime.h>
#include <math.h>

// ---------------------------------------------------------------------------
// GAT (3 layers, 4 heads x 32 ch) + projection for MI455X (gfx1250, wave32).
// Dense GEMMs use V_WMMA_F32_16X16X4_F32 (exact fp32 matrix path).
// Edge phase = 3 passes (atomic max / exp+atomic add / atomic aggregate),
// per-node tables (a_src/a_dst/amax/denom, ~1.6MB each) stay L2-resident.
// All edge-side gathers are float4 (global_load_b128).
// ---------------------------------------------------------------------------

#define HEADS 4
#define CH    32
#define F     128   // HEADS*CH per layer
#define CAT   384   // concat width h0|h1|h2

typedef float v2f __attribute__((ext_vector_type(2)));
typedef float v8f __attribute__((ext_vector_type(8)));

// One wave computes a 16x16 tile of C = A(MxK, ldA) * B(KxN, ldB) [+ bias].
// blockDim.x = 32 * (#col tiles). Requires M % 16 == 0, K % 4 == 0.
__global__ __launch_bounds__(256) void gemm16_wmma(
    const float* __restrict__ A, const float* __restrict__ B,
    float* __restrict__ C, const float* __restrict__ bias,
    int K, int ldA, int ldB, int ldC)
{
    const int lane = threadIdx.x & 31;
    const int wid  = threadIdx.x >> 5;
    const int half = lane >> 4;     // 0: K pair {k,k+1}, 1: {k+2,k+3}
    const int m    = lane & 15;
    const int row0 = blockIdx.x << 4;
    const int col0 = wid << 4;

    const float* Arow = A + (long)(row0 + m) * ldA;
    v8f acc = {};
    for (int k = 0; k < K; k += 4) {
        const int kk = k + (half << 1);
        v2f a, b;
        a.x = Arow[kk];
        a.y = Arow[kk + 1];
        b.x = B[(long)kk       * ldB + col0 + m];
        b.y = B[(long)(kk + 1) * ldB + col0 + m];
        acc = __builtin_amdgcn_wmma_f32_16x16x4_f32(
            false, a, false, b, (short)0, acc, false, false);
    }
    const int   col = col0 + m;
    const float bv  = bias ? bias[col] : 0.0f;
#pragma unroll
    for (int r = 0; r < 8; ++r)
        C[(long)(row0 + r + half * 8) * ldC + col] = acc[r] + bv;
}

// a_src[n,h] = sum_c x[n,h,c]*as[h,c]; a_dst likewise. One thread per (n,h).
__global__ void attn_dots(const float* __restrict__ x,
                          const float* __restrict__ as_, const float* __restrict__ ad_,
                          float* __restrict__ a_src, float* __restrict__ a_dst, int total)
{
    int i = blockIdx.x * blockDim.x + threadIdx.x;
    if (i >= total) return;
    const int h = i & (HEADS - 1);
    const float4* xr = (const float4*)(x + (long)(i >> 2) * F + h * CH);
    const float4* ws = (const float4*)(as_ + h * CH);
    const float4* wd = (const float4*)(ad_ + h * CH);
    float ss = 0.f, sd = 0.f;
#pragma unroll
    for (int c = 0; c < CH / 4; ++c) {
        float4 v = xr[c], s4 = ws[c], d4 = wd[c];
        ss += v.x * s4.x + v.y * s4.y + v.z * s4.z + v.w * s4.w;
        sd += v.x * d4.x + v.y * d4.y + v.z * d4.z + v.w * d4.w;
    }
    a_src[i] = ss;
    a_dst[i] = sd;
}

__global__ void init_nh(float* __restrict__ amax, float* __restrict__ denom, int total)
{
    int i = blockIdx.x * blockDim.x + threadIdx.x;
    if (i >= total) return;
    amax[i]  = -__builtin_inff();
    denom[i] = 0.0f;
}

// out[n, q*4..q*4+3] = bias[...] for this layer's 128-wide slice of N x 384.
// One thread per (n, quad): total = N*32.
__global__ void init_out(float* __restrict__ out, const float* __restrict__ bias, int total)
{
    int i = blockIdx.x * blockDim.x + threadIdx.x;
    if (i >= total) return;
    const int n = i >> 5, q = i & 31;
    float4 bv = ((const float4*)bias)[q];
    *(float4*)(out + (long)n * CAT + q * 4) = bv;
}

__device__ __forceinline__ float leaky(float a) { return a > 0.f ? a : 0.2f * a; }

// Sign-aware float atomic max (int max for >=0, uint min for <0); correct with
// -inf initialization and mixed-sign racing updates.
__device__ __forceinline__ void atomicMaxF(float* addr, float v)
{
    if (v >= 0.f) atomicMax((int*)addr, __float_as_int(v));
    else          atomicMin((unsigned int*)addr, __float_as_uint(v));
}

// One thread per edge: float4 gathers of a_src[s]/a_dst[d], 4 atomic maxes.
__global__ void edge_max(const int* __restrict__ src, const int* __restrict__ dst,
                         const float* __restrict__ a_src, const float* __restrict__ a_dst,
                         float* __restrict__ amax, int total)
{
    int e = blockIdx.x * blockDim.x + threadIdx.x;
    if (e >= total) return;
    const int s = src[e], d = dst[e];
    const float4 as4 = *(const float4*)(a_src + (long)s * HEADS);
    const float4 ad4 = *(const float4*)(a_dst + (long)d * HEADS);
    float* am = amax + (long)d * HEADS;
    atomicMaxF(am + 0, leaky(as4.x + ad4.x));
    atomicMaxF(am + 1, leaky(as4.y + ad4.y));
    atomicMaxF(am + 2, leaky(as4.z + ad4.z));
    atomicMaxF(am + 3, leaky(as4.w + ad4.w));
}

// One thread per edge: recompute alpha, ex = exp(alpha - amax[d]); store ex
// (float4) and accumulate denom with 4 atomic adds.
__global__ void edge_expsum(const int* __restrict__ src, const int* __restrict__ dst,
                            const float* __restrict__ a_src, const float* __restrict__ a_dst,
                            const float* __restrict__ amax,
                            float* __restrict__ exbuf, float* __restrict__ denom, int total)
{
    int e = blockIdx.x * blockDim.x + threadIdx.x;
    if (e >= total) return;
    const int s = src[e], d = dst[e];
    const float4 as4 = *(const float4*)(a_src + (long)s * HEADS);
    const float4 ad4 = *(const float4*)(a_dst + (long)d * HEADS);
    const float4 mx4 = *(const float4*)(amax + (long)d * HEADS);
    float4 ex;
    ex.x = __expf(leaky(as4.x + ad4.x) - mx4.x);
    ex.y = __expf(leaky(as4.y + ad4.y) - mx4.y);
    ex.z = __expf(leaky(as4.z + ad4.z) - mx4.z);
    ex.w = __expf(leaky(as4.w + ad4.w) - mx4.w);
    *(float4*)(exbuf + (long)e * HEADS) = ex;
    float* dn = denom + (long)d * HEADS;
    atomicAdd(dn + 0, ex.x);
    atomicAdd(dn + 1, ex.y);
    atomicAdd(dn + 2, ex.z);
    atomicAdd(dn + 3, ex.w);
}

// One thread per (edge, 4-feature quad q in [0,32)): float4 gather of x[src],
// 4 atomic adds into out[dst]. head = q>>3.
__global__ void edge_agg(const int* __restrict__ src, const int* __restrict__ dst,
                         const float* __restrict__ exbuf, const float* __restrict__ denom,
                         const float* __restrict__ xbuf, float* __restrict__ out, int total)
{
    int i = blockIdx.x * blockDim.x + threadIdx.x;
    if (i >= total) return;
    const int e = i >> 5, q = i & 31;
    const int h = q >> 3;
    const int s = src[e], d = dst[e];
    const float coef = exbuf[(long)e * HEADS + h] /
                       (denom[(long)d * HEADS + h] + 1e-16f);
    const float4 xv = *(const float4*)(xbuf + (long)s * F + q * 4);
    float* o = out + (long)d * CAT + q * 4;
    atomicAdd(o + 0, xv.x * coef);
    atomicAdd(o + 1, xv.y * coef);
    atomicAdd(o + 2, xv.z * coef);
    atomicAdd(o + 3, xv.w * coef);
}

extern "C" void kernel_launch(void* const* d_in, const int* in_sizes, int n_in,
                              void* d_out, int out_size, void* d_ws, size_t ws_size,
                              hipStream_t stream)
{
    const int N = in_sizes[0] / F;      // 100000
    const int E = in_sizes[1] / 2;      // 1600000

    const float* x   = (const float*)d_in[0];
    const int*   ei  = (const int*)d_in[1];
    const int*   src = ei;
    const int*   dst = ei + E;
    const float* W[3]  = { (const float*)d_in[2],  (const float*)d_in[6],  (const float*)d_in[10] };
    const float* AS[3] = { (const float*)d_in[3],  (const float*)d_in[7],  (const float*)d_in[11] };
    const float* AD[3] = { (const float*)d_in[4],  (const float*)d_in[8],  (const float*)d_in[12] };
    const float* Bs[3] = { (const float*)d_in[5],  (const float*)d_in[9],  (const float*)d_in[13] };
    const float* Wp = (const float*)d_in[14];
    const float* bp = (const float*)d_in[15];

    // workspace layout (floats)
    float* ws    = (float*)d_ws;
    float* hcat  = ws;                                // N*384 concat [h0|h1|h2]
    float* xbuf  = hcat + (size_t)N * CAT;            // N*128 pre-agg features
    float* asrc  = xbuf + (size_t)N * F;              // N*4
    float* adst  = asrc + (size_t)N * HEADS;          // N*4
    float* amax  = adst + (size_t)N * HEADS;          // N*4
    float* denom = amax + (size_t)N * HEADS;          // N*4
    float* exb   = denom + (size_t)N * HEADS;         // E*4

    const int nh  = N * HEADS;          // per-(node,head) work
    const int nq  = N * 32;             // per-(node,quad) work
    const int eq  = E * 32;             // per-(edge,quad) work
    const int gNH = (nh + 255) / 256;
    const int gNQ = (nq + 255) / 256;
    const int gE  = (E + 255) / 256;
    const int gEQ = (eq + 255) / 256;

    for (int l = 0; l < 3; ++l) {
        const float* inA = (l == 0) ? x : (hcat + (size_t)(l - 1) * F);
        const int ldA = (l == 0) ? F : CAT;

        // x_l = h_{l-1} @ W_l   (N x 128) — fp32 WMMA, 8 col-tiles per block
        gemm16_wmma<<<N / 16, 256, 0, stream>>>(inA, W[l], xbuf, nullptr, F, ldA, F, F);
        attn_dots<<<gNH, 256, 0, stream>>>(xbuf, AS[l], AD[l], asrc, adst, nh);
        init_nh<<<gNH, 256, 0, stream>>>(amax, denom, nh);
        init_out<<<gNQ, 256, 0, stream>>>(hcat + (size_t)l * F, Bs[l], nq);
        edge_max<<<gE, 256, 0, stream>>>(src, dst, asrc, adst, amax, E);
        edge_expsum<<<gE, 256, 0, stream>>>(src, dst, asrc, adst, amax, exb, denom, E);
        edge_agg<<<gEQ, 256, 0, stream>>>(src, dst, exb, denom, xbuf,
                                          hcat + (size_t)l * F, eq);
    }

    // out = [h0|h1|h2] @ Wp + bp   (N x 32) — 2 col-tiles per block
    gemm16_wmma<<<N / 16, 64, 0, stream>>>(hcat, Wp, (float*)d_out, bp, CAT, CAT, 32, 32);
}